// Rudy_16561393894026
// MI455X (gfx1250) — compile-verified
//
#include <hip/hip_runtime.h>
#include <hip/hip_bf16.h>

// ---------------------------------------------------------------------------
// RUDY congestion maps on gfx1250.
// einsum "nx,ny->xy" == 256x256 GEMM with K = num_nets (100k), computed in
// fp32 with V_WMMA_F32_16X16X4_F32 (exact reference math, fp32 accumulate).
// Each wave owns a 16x128 output strip (8 ty-tiles x both maps = 16 WMMAs
// per K-group) so the VALU operand construction is amortized ~7x.
// ---------------------------------------------------------------------------

typedef __attribute__((ext_vector_type(2))) float v2f;
typedef __attribute__((ext_vector_type(8))) float v8f;

#define NBINS    256
#define BIN_SZ   2.0f
// fold 1/(bin_area * UNIT_CAP) into the per-net weights
#define INV_HCAP (1.0f / (4.0f * 50.0f))
#define INV_VCAP (1.0f / (4.0f * 40.0f))
#define KCHUNK   256    // net-groups (of 4 nets) per wave
#define TTILES   8      // ty tiles per wave (16x128 strip)

// ---------------------------------------------------------------------------
// Pass 1: per-net bounding box + scaled RUDY weights -> 8 floats per net.
// Padded nets (n >= nNets) get all-zero records (zero weight => no deposit).
// ---------------------------------------------------------------------------
__global__ void rudy_bbox(const float* __restrict__ pin_pos,
                          const int*   __restrict__ netpin_start,
                          const int*   __restrict__ flat_netpin,
                          const float* __restrict__ net_w,
                          float* __restrict__ nets,
                          int nNets, int nPad)
{
    int n = blockIdx.x * blockDim.x + threadIdx.x;
    if (n >= nPad) return;

    float xmin = 0.f, xmax = 0.f, ymin = 0.f, ymax = 0.f;
    float wth = 0.f, wtv = 0.f;
    if (n < nNets) {
        int s = netpin_start[n];
        int e = netpin_start[n + 1];
        int p0 = flat_netpin[s];
        xmin = xmax = pin_pos[2 * p0];
        ymin = ymax = pin_pos[2 * p0 + 1];
        for (int j = s + 1; j < e; ++j) {
            int p = flat_netpin[j];
            float x = pin_pos[2 * p];
            float y = pin_pos[2 * p + 1];
            xmin = fminf(xmin, x); xmax = fmaxf(xmax, x);
            ymin = fminf(ymin, y); ymax = fmaxf(ymax, y);
        }
        float w = net_w[n];
        wth = w / (ymax - ymin) * INV_HCAP;   // horizontal demand density
        wtv = w / (xmax - xmin) * INV_VCAP;   // vertical demand density
    }
    float4* o = (float4*)(nets + (size_t)n * 8);
    o[0] = make_float4(xmin, xmax, ymin, ymax);
    o[1] = make_float4(wth, wtv, 0.f, 0.f);
}

// ---------------------------------------------------------------------------
// Pass 2: seed H/V output maps with the initial utilization maps.
// ---------------------------------------------------------------------------
__global__ void rudy_init(const float* __restrict__ ih,
                          const float* __restrict__ iv,
                          float* __restrict__ H, float* __restrict__ V, int n)
{
    int i = blockIdx.x * blockDim.x + threadIdx.x;
    if (i < n) { H[i] = ih[i]; V[i] = iv[i]; }
}

// ---------------------------------------------------------------------------
// Pass 3: the GEMM.  One wave per (16x128 output strip, K-chunk).
// blockDim = 32 keeps the group index g wave-uniform -> net records load as
// scalar (SMEM) loads; each lane-half owns nets (2h, 2h+1) of the group, which
// matches BOTH the f32 A(16x4) and B(4x16) WMMA lane layouts.
//   A[M][k] = ox(net_k, bin_x = tx*16 + M) * wt     (M = lane&15, k in v0,v1)
//   B[k][N] = oy(net_k, bin_y)                      (N = lane&15)
//   C/D[M][N]: VGPR r <-> M = r + 8*(lane>=16), N = lane&15
// Per K-group: build A (+weights) once, then 8 cheap B rebuilds sweep the
// 128-bin y-strip -> 16 WMMAs per ~88 VALU.
// ---------------------------------------------------------------------------
__global__ __launch_bounds__(32) void rudy_gemm(const float* __restrict__ nets,
                                                float* __restrict__ Hout,
                                                float* __restrict__ Vout,
                                                int nGroups)
{
    const int lane  = threadIdx.x;          // 0..31
    const int wid   = blockIdx.x;
    const int strip = wid & 31;             // 16 tx-tiles x 2 y-strips
    const int chunk = wid >> 5;
    const int tx = strip >> 1;              // which 16-row block of bin-x
    const int sy = strip & 1;               // which 128-col strip of bin-y
    const int h  = lane >> 4;               // lane-half selects net pair
    const int l  = lane & 15;
    const bool hi = (h != 0);

    const float lox  = (float)(tx * 16 + l) * BIN_SZ;   // bin-x low edge (A)
    const float loy0 = (float)(sy * 128 + l) * BIN_SZ;  // base bin-y low edge

    v8f cH[TTILES], cV[TTILES];
#pragma unroll
    for (int t = 0; t < TTILES; ++t) {
        cH[t] = (v8f){0.f, 0.f, 0.f, 0.f, 0.f, 0.f, 0.f, 0.f};
        cV[t] = (v8f){0.f, 0.f, 0.f, 0.f, 0.f, 0.f, 0.f, 0.f};
    }

    int g0 = chunk * KCHUNK;
    int g1 = g0 + KCHUNK;
    if (g1 > nGroups) g1 = nGroups;

    for (int g = g0; g < g1; ++g) {
        // 4 nets * {bbox, weights} -- wave-uniform address => scalar loads
        const float4* nd = (const float4*)nets + (size_t)g * 8;
        float4 b0 = nd[0], w0 = nd[1];      // net 0
        float4 b1 = nd[2], w1 = nd[3];      // net 1
        float4 b2 = nd[4], w2 = nd[5];      // net 2
        float4 b3 = nd[6], w3 = nd[7];      // net 3

        // this lane's two nets: (2h, 2h+1)
        float xmn0 = hi ? b2.x : b0.x, xmx0 = hi ? b2.y : b0.y;
        float ymn0 = hi ? b2.z : b0.z, ymx0 = hi ? b2.w : b0.w;
        float wh0  = hi ? w2.x : w0.x, wv0  = hi ? w2.y : w0.y;
        float xmn1 = hi ? b3.x : b1.x, xmx1 = hi ? b3.y : b1.y;
        float ymn1 = hi ? b3.z : b1.z, ymx1 = hi ? b3.w : b1.w;
        float wh1  = hi ? w3.x : w1.x, wv1  = hi ? w3.y : w1.y;

        // A tile: clipped x-overlaps, weighted (built once per K-group)
        float ox0 = fmaxf(fminf(xmx0, lox + BIN_SZ) - fmaxf(xmn0, lox), 0.f);
        float ox1 = fmaxf(fminf(xmx1, lox + BIN_SZ) - fmaxf(xmn1, lox), 0.f);
        v2f aH = { ox0 * wh0, ox1 * wh1 };
        v2f aV = { ox0 * wv0, ox1 * wv1 };

        // sweep 8 ty-tiles: cheap B rebuild (8 VALU) + 2 WMMAs each
#pragma unroll
        for (int t = 0; t < TTILES; ++t) {
            float loy = loy0 + (float)(t * 16) * BIN_SZ;
            float oy0 = fmaxf(fminf(ymx0, loy + BIN_SZ) - fmaxf(ymn0, loy), 0.f);
            float oy1 = fmaxf(fminf(ymx1, loy + BIN_SZ) - fmaxf(ymn1, loy), 0.f);
            v2f bb = { oy0, oy1 };
            cH[t] = __builtin_amdgcn_wmma_f32_16x16x4_f32(
                        false, aH, false, bb, (short)0, cH[t], false, false);
            cV[t] = __builtin_amdgcn_wmma_f32_16x16x4_f32(
                        false, aV, false, bb, (short)0, cV[t], false, false);
        }
    }

    // C/D layout: VGPR r -> M = r + 8*h ; N = lane&15
    const int row0 = tx * 16 + h * 8;
#pragma unroll
    for (int t = 0; t < TTILES; ++t) {
        const int col = sy * 128 + t * 16 + l;
#pragma unroll
        for (int r = 0; r < 8; ++r) {
            int idx = (row0 + r) * NBINS + col;
            atomicAdd(&Hout[idx], cH[t][r]);
            atomicAdd(&Vout[idx], cV[t][r]);
        }
    }
}

// ---------------------------------------------------------------------------
// Pass 4: route = max(|H|, |V|)
// ---------------------------------------------------------------------------
__global__ void rudy_route(const float* __restrict__ H,
                           const float* __restrict__ V,
                           float* __restrict__ route, int n)
{
    int i = blockIdx.x * blockDim.x + threadIdx.x;
    if (i < n) route[i] = fmaxf(fabsf(H[i]), fabsf(V[i]));
}

// ---------------------------------------------------------------------------
extern "C" void kernel_launch(void* const* d_in, const int* in_sizes, int n_in,
                              void* d_out, int out_size, void* d_ws, size_t ws_size,
                              hipStream_t stream)
{
    const float* pin_pos      = (const float*)d_in[0];
    const int*   netpin_start = (const int*)  d_in[1];
    const int*   flat_netpin  = (const int*)  d_in[2];
    const float* net_w        = (const float*)d_in[3];
    const float* init_h       = (const float*)d_in[4];
    const float* init_v       = (const float*)d_in[5];

    const int nNets   = in_sizes[3];
    const int nGroups = (nNets + 3) / 4;        // 4 nets per WMMA K-step
    const int nPad    = nGroups * 4;
    const int nChunks = (nGroups + KCHUNK - 1) / KCHUNK;
    const int nb      = NBINS * NBINS;

    float* nets  = (float*)d_ws;                // nPad * 8 floats (~3.2 MB)
    float* route = (float*)d_out;               // [256,256]
    float* H     = route + nb;                  // [256,256]
    float* V     = H + nb;                      // [256,256]

    rudy_bbox<<<(nPad + 255) / 256, 256, 0, stream>>>(
        pin_pos, netpin_start, flat_netpin, net_w, nets, nNets, nPad);

    rudy_init<<<(nb + 255) / 256, 256, 0, stream>>>(init_h, init_v, H, V, nb);

    // 32 output strips x nChunks K-chunks, one wave (32 threads) each
    rudy_gemm<<<32 * nChunks, 32, 0, stream>>>(nets, H, V, nGroups);

    rudy_route<<<(nb + 255) / 256, 256, 0, stream>>>(H, V, route, nb);
}